// FuzzyARTMAP_46643344835326
// MI455X (gfx1250) — compile-verified
//
#include <hip/hip_runtime.h>
#include <stdint.h>

// Fuzzy ART match on MI455X (gfx1250, wave32).
// match[n,c] = sum_d min(x[n,d], w[c,d]) / sum_d x[n,d]; vigilance gate; row argmax.
// Dominant op is a min-plus reduction -> no ring-matmul factorization exists, so the
// bulk runs on packed VALU out of LDS. CDNA5-specific paths used:
//   * global_load_async_to_lds_b128 (ASYNCcnt) for tile staging
//   * v_wmma_f32_16x16x4_f32 accumulator chain (A = x-tile, B = ones) for the exact
//     f32 row sums sum_d x[n,d] (a real matmul: x @ ones)

#define VIGILANCE 0.75f
#define D_DIM     256
#define TILE_N    32
#define TILE_C    64
#define STRIDE    260   // 256 + 4-float pad: row-to-row bank offset = 4 -> conflict-free
#define THREADS   256

#if defined(__gfx1250__) && defined(__AMDGCN__) && \
    __has_builtin(__builtin_amdgcn_global_load_async_to_lds_b128) && \
    __has_builtin(__builtin_amdgcn_s_wait_asynccnt)
#define USE_ASYNC_LDS 1
#else
#define USE_ASYNC_LDS 0
#endif

#if defined(__gfx1250__) && __has_builtin(__builtin_amdgcn_wmma_f32_16x16x4_f32)
#define USE_WMMA_ROWSUM 1
#else
#define USE_WMMA_ROWSUM 0
#endif

typedef int   v4i __attribute__((ext_vector_type(4)));
typedef float v2f __attribute__((ext_vector_type(2)));
typedef float v8f __attribute__((ext_vector_type(8)));
typedef __attribute__((address_space(1))) v4i* as1_v4i_ptr;
typedef __attribute__((address_space(3))) v4i* as3_v4i_ptr;

__device__ __forceinline__ void copy16_g2l(const float* __restrict__ g, float* l) {
#if USE_ASYNC_LDS
  // Global side: AS(1) shares the generic bit pattern -> integer cast is safe (drops const).
  // LDS side: direct pointer cast -> addrspacecast (generic LDS ptrs carry aperture bits).
  __builtin_amdgcn_global_load_async_to_lds_b128(
      (as1_v4i_ptr)(uintptr_t)g,
      (as3_v4i_ptr)l,
      /*offset=*/0, /*cpol=*/0);
#else
  *(float4*)l = *(const float4*)g;
#endif
}

__global__ __launch_bounds__(THREADS)
void fuzzy_art_match_kernel(const float* __restrict__ x,
                            const float* __restrict__ w,
                            float* __restrict__ out, int N, int C) {
  extern __shared__ float lds[];
  float* ldsX = lds;                              // TILE_N * STRIDE floats
  float* ldsW = lds + TILE_N * STRIDE;            // TILE_C * STRIDE floats
  float* invs = ldsW + TILE_C * STRIDE;           // TILE_N floats

  const int t      = threadIdx.x;
  const int c_base = blockIdx.x * TILE_C;
  const int n_base = blockIdx.y * TILE_N;

  // ---- Stage tiles into LDS via async data mover (tracked by ASYNCcnt) ----
  for (int q = t; q < TILE_N * (D_DIM / 4); q += THREADS) {
    const int row = q >> 6;
    const int col = (q & 63) << 2;
    copy16_g2l(&x[(size_t)(n_base + row) * D_DIM + col], &ldsX[row * STRIDE + col]);
  }
  for (int q = t; q < TILE_C * (D_DIM / 4); q += THREADS) {
    const int row = q >> 6;
    const int col = (q & 63) << 2;
    copy16_g2l(&w[(size_t)(c_base + row) * D_DIM + col], &ldsW[row * STRIDE + col]);
  }
#if USE_ASYNC_LDS
  __builtin_amdgcn_s_wait_asynccnt(0);   // s_wait_asynccnt 0: this wave's DMA done
#endif
  __syncthreads();                        // make all waves' LDS writes visible

  // ---- Per-row 1/sum(x): exact f32 matmul x_tile @ ones via V_WMMA_F32_16X16X4_F32 ----
#if USE_WMMA_ROWSUM
  {
    const int wave = t >> 5;
    const int lane = t & 31;
    if (wave < 2) {                       // wave-uniform branch: EXEC all-1s inside (WMMA req)
      const int hi = lane >> 4;           // 0 -> K pair {k,k+1}; 1 -> K pair {k+2,k+3}
      const int m  = (wave << 4) + (lane & 15);   // this lane's x row (16 rows per wave)
      const float* xrow = &ldsX[(size_t)m * STRIDE];
      v2f onesb; onesb.x = 1.0f; onesb.y = 1.0f;  // B = 4x16 all-ones
      v8f acc = {};
      for (int k = 0; k < D_DIM; k += 4) {
        v2f a;
        a.x = xrow[k + 2 * hi];           // A 16x4 layout: lanes 0-15 K{0,1}, lanes 16-31 K{2,3}
        a.y = xrow[k + 2 * hi + 1];
        acc = __builtin_amdgcn_wmma_f32_16x16x4_f32(
            /*neg_a=*/false, a, /*neg_b=*/false, onesb,
            /*c_mod=*/(short)0, acc, /*reuse_a=*/false, /*reuse_b=*/false);
      }
      // D layout: VGPR r, lanes 0-15 hold M=r; lanes 16-31 hold M=r+8 (same value over N)
      if ((lane & 15) == 0) {
#pragma unroll
        for (int r = 0; r < 8; ++r)
          invs[(wave << 4) + (hi << 3) + r] = 1.0f / acc[r];
      }
    }
  }
#else
  if (t < TILE_N) {
    float s = 0.f;
    for (int d = 0; d < D_DIM; d += 4) {
      float4 v = *(const float4*)&ldsX[t * STRIDE + d];
      s += v.x + v.y + v.z + v.w;
    }
    invs[t] = 1.0f / s;
  }
#endif
  __syncthreads();

  // ---- Min-plus tile compute: each thread owns 2 n-rows x 4 c-cols ----
  const int cid = t & 15;   // 0..15 -> c = cid + 16*j  (16 distinct W rows/wave -> 64 banks)
  const int nid = t >> 4;   // 0..15 -> n = 2*nid + {0,1} (X rows broadcast across lanes)
  const int n0  = nid * 2;

  float4 acc0[4], acc1[4];
#pragma unroll
  for (int j = 0; j < 4; ++j) {
    acc0[j] = make_float4(0.f, 0.f, 0.f, 0.f);
    acc1[j] = make_float4(0.f, 0.f, 0.f, 0.f);
  }

  const float* xr0 = &ldsX[(size_t)n0 * STRIDE];
  const float* xr1 = &ldsX[(size_t)(n0 + 1) * STRIDE];

#pragma unroll 4
  for (int d = 0; d < D_DIM; d += 4) {
    const float4 xv0 = *(const float4*)&xr0[d];
    const float4 xv1 = *(const float4*)&xr1[d];
#pragma unroll
    for (int j = 0; j < 4; ++j) {
      const float4 wv = *(const float4*)&ldsW[(size_t)(cid + 16 * j) * STRIDE + d];
      acc0[j].x += fminf(xv0.x, wv.x);
      acc0[j].y += fminf(xv0.y, wv.y);
      acc0[j].z += fminf(xv0.z, wv.z);
      acc0[j].w += fminf(xv0.w, wv.w);
      acc1[j].x += fminf(xv1.x, wv.x);
      acc1[j].y += fminf(xv1.y, wv.y);
      acc1[j].z += fminf(xv1.z, wv.z);
      acc1[j].w += fminf(xv1.w, wv.w);
    }
  }

  const float inv0 = invs[n0];
  const float inv1 = invs[n0 + 1];
#pragma unroll
  for (int j = 0; j < 4; ++j) {
    const int cg = c_base + cid + 16 * j;
    float m0 = (acc0[j].x + acc0[j].y + acc0[j].z + acc0[j].w) * inv0;
    float m1 = (acc1[j].x + acc1[j].y + acc1[j].z + acc1[j].w) * inv1;
    m0 = (m0 >= VIGILANCE) ? m0 : 0.0f;
    m1 = (m1 >= VIGILANCE) ? m1 : 0.0f;
    out[(size_t)(n_base + n0)     * C + cg] = m0;
    out[(size_t)(n_base + n0 + 1) * C + cg] = m1;
  }
}

// One wave32 per row: coalesced strided scan + butterfly argmax (first-max index).
__global__ __launch_bounds__(256)
void fuzzy_art_argmax_kernel(const float* __restrict__ scores,
                             float* __restrict__ out_idx, int N, int C) {
  const int wave = threadIdx.x >> 5;
  const int lane = threadIdx.x & 31;
  const int row  = blockIdx.x * 8 + wave;
  if (row >= N) return;

  const float* r = scores + (size_t)row * C;
  float bv = -1.0f;
  int   bi = 0;
  for (int c = lane; c < C; c += 32) {
    const float v = r[c];
    if (v > bv) { bv = v; bi = c; }   // strict > : earliest index within lane
  }
  for (int off = 16; off >= 1; off >>= 1) {
    const float ov = __shfl_xor(bv, off, 32);
    const int   oi = __shfl_xor(bi, off, 32);
    if (ov > bv || (ov == bv && oi < bi)) { bv = ov; bi = oi; }
  }
  if (lane == 0) out_idx[row] = (float)bi;
}

extern "C" void kernel_launch(void* const* d_in, const int* in_sizes, int n_in,
                              void* d_out, int out_size, void* d_ws, size_t ws_size,
                              hipStream_t stream) {
  const float* x = (const float*)d_in[0];   // [N, 256]
  const float* w = (const float*)d_in[1];   // [C, 256]
  float* out = (float*)d_out;               // [N*C scores | N indices-as-float]

  const int N = in_sizes[0] / D_DIM;        // 1024
  const int C = in_sizes[1] / D_DIM;        // 512

  dim3 grid(C / TILE_C, N / TILE_N);        // 8 x 32 = 256 blocks
  const size_t sh_bytes =
      (size_t)(TILE_N * STRIDE + TILE_C * STRIDE + TILE_N) * sizeof(float); // ~98 KB < 320 KB WGP LDS
  fuzzy_art_match_kernel<<<grid, THREADS, sh_bytes, stream>>>(x, w, out, N, C);

  float* idx_out = out + (size_t)N * C;
  fuzzy_art_argmax_kernel<<<dim3((N + 7) / 8), 256, 0, stream>>>(out, idx_out, N, C);
}